// StateStack_91242285236581
// MI455X (gfx1250) — compile-verified
//
#include <hip/hip_runtime.h>
#include <hip/hip_bf16.h>

// Problem constants (fixed by the reference file).
constexpr int T = 64;
constexpr int B = 512;
constexpr int H = 512;
constexpr int H4 = H / 4;  // float4 chunks per row = 128

// GCC-style int 4-vector: matches the builtin's expected pointee type
// (diagnostic showed: __attribute__((__vector_size__(4 * sizeof(int)))) int).
typedef int v4i __attribute__((vector_size(16)));

// -----------------------------------------------------------------------------
// Kernel 1: per-batch integer scan over ops.
// hidden[t,b] = inputs[src,b,:] where src is the last timestep whose write
// slot (pos_before+1) equals this step's read slot (pos_after); src = -1
// means read slot 0 => initial_hidden.
// -----------------------------------------------------------------------------
__global__ void statestack_scan_kernel(const int* __restrict__ ops,
                                       int* __restrict__ src) {
  const int b = blockIdx.x * blockDim.x + threadIdx.x;
  if (b >= B) return;

  int lw[T + 2];  // last writer per stack slot (slots 1..T used)
#pragma unroll
  for (int i = 0; i < T + 2; ++i) lw[i] = -1;

  int pos = 0;
  for (int t = 0; t < T; ++t) {
    const int op = ops[t * B + b];  // coalesced across b for fixed t
    lw[pos + 1] = t;                // write x[t] at slot pos+1
    pos += op;                      // pos stays >= 0 by construction
    src[t * B + b] = (pos == 0) ? -1 : lw[pos];
  }
}

// -----------------------------------------------------------------------------
// Kernel 2: row gather. One block per output row (t,b); 128 threads each move
// one 16B chunk. Primary path: CDNA5 async global->LDS->global copy
// (no VGPR round trip, tracked by ASYNCcnt). Fallback: plain float4 copy.
// -----------------------------------------------------------------------------
#if defined(__HIP_DEVICE_COMPILE__) &&                                   \
    __has_builtin(__builtin_amdgcn_global_load_async_to_lds_b128) &&     \
    __has_builtin(__builtin_amdgcn_global_store_async_from_lds_b128)
#define STATESTACK_USE_ASYNC_LDS 1
#else
#define STATESTACK_USE_ASYNC_LDS 0
#endif

__global__ void statestack_gather_kernel(const float* __restrict__ inputs,
                                         const float* __restrict__ init,
                                         const int* __restrict__ srcmap,
                                         float* __restrict__ out) {
  const int row = blockIdx.x;       // row = t*B + b
  const int b = row & (B - 1);      // B = 512 (power of two)
  const int s = srcmap[row];        // uniform per block -> scalar load

  const float4* src4 =
      (s < 0) ? reinterpret_cast<const float4*>(init)
              : reinterpret_cast<const float4*>(inputs +
                                                ((size_t)s * B + b) * (size_t)H);
  float4* dst4 = reinterpret_cast<float4*>(out + (size_t)row * (size_t)H);

  const int i = threadIdx.x;  // 0..H4-1

#if STATESTACK_USE_ASYNC_LDS
  __shared__ float4 buf[H4];  // 2 KB staging per row

  // global -> LDS (async, 16B per lane)
  __builtin_amdgcn_global_load_async_to_lds_b128(
      (__attribute__((address_space(1))) v4i*)(src4 + i),
      (__attribute__((address_space(3))) v4i*)(&buf[i]),
      0, 0);

  // Each lane stores the same 16B slot it loaded: no cross-lane sharing, so
  // only the async counter needs to drain (no workgroup barrier required).
#if __has_builtin(__builtin_amdgcn_s_wait_asynccnt)
  __builtin_amdgcn_s_wait_asynccnt(0);
#else
  asm volatile("s_wait_asynccnt 0" ::: "memory");
#endif

  // LDS -> global (async). S_ENDPGM's implicit wait-idle drains the store.
  __builtin_amdgcn_global_store_async_from_lds_b128(
      (__attribute__((address_space(1))) v4i*)(dst4 + i),
      (__attribute__((address_space(3))) v4i*)(&buf[i]),
      0, 0);
#else
  dst4[i] = src4[i];
#endif
}

extern "C" void kernel_launch(void* const* d_in, const int* in_sizes, int n_in,
                              void* d_out, int out_size, void* d_ws,
                              size_t ws_size, hipStream_t stream) {
  (void)in_sizes;
  (void)n_in;
  (void)out_size;
  (void)ws_size;

  const float* inputs = (const float*)d_in[0];  // (T, B, H) fp32
  const float* init   = (const float*)d_in[1];  // (H,) fp32
  const int*   ops    = (const int*)d_in[2];    // (T, B) int32
  // d_in[3] = seq_len (unused: pos <= T <= SEQ always)

  float* out = (float*)d_out;        // (T, B, H) fp32
  int* srcmap = (int*)d_ws;          // T*B ints = 128 KB scratch

  statestack_scan_kernel<<<(B + 255) / 256, 256, 0, stream>>>(ops, srcmap);
  statestack_gather_kernel<<<T * B, H4, 0, stream>>>(inputs, init, srcmap, out);
}